// Seasonal_Quantum_Engine_47167330844888
// MI455X (gfx1250) — compile-verified
//
#include <hip/hip_runtime.h>
#include <hip/hip_bf16.h>

// ---------------------------------------------------------------------------
// Problem constants (match reference)
// ---------------------------------------------------------------------------
#define NQ   10
#define NL   4
#define QDIM 1024      // 2^NQ
#define BSZ  4096
#define SEQL 512
#define PREDN 96

typedef float v2f __attribute__((ext_vector_type(2)));
typedef float v8f __attribute__((ext_vector_type(8)));

__device__ __forceinline__ v8f sqe_wmma(v2f a, v2f b, v8f c) {
    return __builtin_amdgcn_wmma_f32_16x16x4_f32(
        /*neg_a=*/false, a, /*neg_b=*/false, b,
        /*c_mod=*/(short)0, c, /*reuse_a=*/false, /*reuse_b=*/false);
}

// ---------------------------------------------------------------------------
// Kernel 1: h = x @ W_in^T + b_in + 1e-6   (fp32 WMMA, 16x16x4)
//
// One wave32 owns a 32x32 C macro-tile (2x2 grid of 16x16 WMMA tiles), so each
// A fragment is reused by 2 WMMAs and each B fragment by 2 WMMAs:
// 4 global_load_b64 per 4 v_wmma (1 load/WMMA, vs 2 with a single tile).
//
// Fragment layouts per CDNA5 ISA §7.12.2 (f32):
//   A (16x4, 2 VGPR): lane 0-15 -> M=lane, K = k+{0,1}; lane 16-31 -> K = k+{2,3}
//   B (4x16, 2 VGPR): lane 0-15 -> N=lane, same K split
//   -> each lane's two K values are contiguous: one float2 (b64) load/operand.
//   C (16x16, 8 VGPR): vgpr r, lane l -> M = r + 8*(l>=16), N = l%16
// ---------------------------------------------------------------------------
__global__ __launch_bounds__(256)
void sqe_gemm_in(const float* __restrict__ x,
                 const float* __restrict__ W_in,
                 const float* __restrict__ b_in,
                 float* __restrict__ h)
{
    const int wave = blockIdx.x * (blockDim.x >> 5) + (threadIdx.x >> 5);
    const int lane = threadIdx.x & 31;
    const int tilesN2 = QDIM / 32;            // 32 macro-tiles along N
    const int tm = wave / tilesN2;            // 0..127
    const int tn = wave % tilesN2;            // 0..31
    const int lm = lane & 15;
    const int hi = lane >> 4;                 // selects K pair {0,1} vs {2,3}

    const float* a0p = x    + (size_t)(tm * 32 + lm) * SEQL + 2 * hi;
    const float* a1p = a0p  + (size_t)16 * SEQL;
    const float* b0p = W_in + (size_t)(tn * 32 + lm) * SEQL + 2 * hi;
    const float* b1p = b0p  + (size_t)16 * SEQL;

    v8f acc00 = {}, acc01 = {}, acc10 = {}, acc11 = {};
    #pragma unroll 2
    for (int k = 0; k < SEQL; k += 4) {
        const v2f A0 = *(const v2f*)(a0p + k);
        const v2f A1 = *(const v2f*)(a1p + k);
        const v2f B0 = *(const v2f*)(b0p + k);
        const v2f B1 = *(const v2f*)(b1p + k);
        acc00 = sqe_wmma(A0, B0, acc00);
        acc01 = sqe_wmma(A0, B1, acc01);
        acc10 = sqe_wmma(A1, B0, acc10);
        acc11 = sqe_wmma(A1, B1, acc11);
    }

    const float bias0 = b_in[tn * 32 + lm]      + 1e-6f;   // N-tile 0 bias
    const float bias1 = b_in[tn * 32 + 16 + lm] + 1e-6f;   // N-tile 1 bias

    float* o0 = h + (size_t)(tm * 32 + 8 * hi) * QDIM + tn * 32 + lm;
    float* o1 = o0 + (size_t)16 * QDIM;
    #pragma unroll
    for (int r = 0; r < 8; ++r) {
        o0[(size_t)r * QDIM]      = acc00[r] + bias0;
        o0[(size_t)r * QDIM + 16] = acc01[r] + bias1;
        o1[(size_t)r * QDIM]      = acc10[r] + bias0;
        o1[(size_t)r * QDIM + 16] = acc11[r] + bias1;
    }
}

// ---------------------------------------------------------------------------
// Layer permutation: PERMS[l][i] = g_0(g_1(...g_9(i))) where
// g_c(s) = s ^ (((s >> (NQ-1-c)) & 1) << (NQ-1-(c+r)%NQ)),  r = l%(NQ-1)+1
// ---------------------------------------------------------------------------
__device__ __forceinline__ int sqe_perm_idx(int l, int i)
{
    const int r = l % (NQ - 1) + 1;
    int s = i;
    #pragma unroll
    for (int c = NQ - 1; c >= 0; --c) {
        const int t   = (c + r) % NQ;
        const int bit = (s >> (NQ - 1 - c)) & 1;
        s ^= bit << (NQ - 1 - t);
    }
    return s;
}

// ---------------------------------------------------------------------------
// Kernel 2: per-row circuit. 1 block = 1 batch row, 256 threads (8 wave32).
// The h row is brought into LDS with GLOBAL_LOAD_ASYNC_TO_LDS_B128 (each lane
// supplies its LDS dest offset + global address), overlapped with the gate-
// matrix sincos work, then fenced with s_wait_asynccnt.
// psi (1024 complex = 8 KB) stays in LDS for all 40 gates + 4 permutes.
// ---------------------------------------------------------------------------
__global__ __launch_bounds__(256)
void sqe_circuit(const float* __restrict__ h,
                 const float* __restrict__ qw,     // (NL, NQ, 3)
                 const float* __restrict__ W_out,  // (PREDN, NQ)
                 const float* __restrict__ b_out,  // (PREDN)
                 float* __restrict__ out)          // (BSZ, PREDN)
{
    __shared__ float  s_stage[QDIM];      // async-copy landing buffer (4 KB)
    __shared__ float2 s_psi[QDIM];        // 8 KB state
    __shared__ float  s_g[NL * NQ][8];    // 40 gates x {m00,m01,m10,m11} (re,im)
    __shared__ float  s_red[8];
    __shared__ float  s_wq[8][NQ];
    __shared__ float  s_exp[NQ];
    __shared__ float  s_inv;

    const int tid  = threadIdx.x;
    const int lane = tid & 31;
    const int wv   = tid >> 5;
    const int brow = blockIdx.x;
    const float* hrow = h + (size_t)brow * QDIM;

    // --- kick off async global->LDS copy of the h row (16 B per lane) ---
    // LDS aperture occupies flat-addr[63:32]; low 32 bits are the LDS address.
    {
        const unsigned lds_dst = (unsigned)(uintptr_t)(&s_stage[tid * 4]);
        const float*   gsrc    = hrow + tid * 4;
        asm volatile("global_load_async_to_lds_b128 %0, %1, off"
                     :: "v"(lds_dst), "v"(gsrc) : "memory");
    }

    // --- gate matrices (threads 0..39) overlap the async copy ---
    if (tid < NL * NQ) {
        const float* wgt = qw + tid * 3;
        const float phi = wgt[0], theta = wgt[1], omega = wgt[2];
        float st, ct;  __sincosf(0.5f * theta, &st, &ct);
        float sa, ca;  __sincosf(0.5f * (phi + omega), &sa, &ca);
        float sb, cb;  __sincosf(0.5f * (phi - omega), &sb, &cb);
        // m00 = conj(ep)*c ; m01 = -em*s ; m10 = conj(em)*s ; m11 = ep*c
        s_g[tid][0] =  ca * ct;  s_g[tid][1] = -sa * ct;
        s_g[tid][2] = -cb * st;  s_g[tid][3] = -sb * st;
        s_g[tid][4] =  cb * st;  s_g[tid][5] = -sb * st;
        s_g[tid][6] =  ca * ct;  s_g[tid][7] =  sa * ct;
    }

    // --- fence the async copy, then everyone sees the row in LDS ---
    asm volatile("s_wait_asynccnt 0x0" ::: "memory");
    __syncthreads();

    // --- sum of squares (wave32 shuffle reduce, then LDS) ---
    const float4 hv = *(const float4*)(&s_stage[tid * 4]);
    float ss = hv.x * hv.x + hv.y * hv.y + hv.z * hv.z + hv.w * hv.w;
    #pragma unroll
    for (int off = 16; off; off >>= 1) ss += __shfl_xor(ss, off, 32);
    if (lane == 0) s_red[wv] = ss;
    __syncthreads();
    if (tid == 0) {
        float t = 0.f;
        #pragma unroll
        for (int i = 0; i < 8; ++i) t += s_red[i];
        s_inv = rsqrtf(t);
    }
    __syncthreads();
    const float inv = s_inv;
    s_psi[tid * 4 + 0] = make_float2(hv.x * inv, 0.f);
    s_psi[tid * 4 + 1] = make_float2(hv.y * inv, 0.f);
    s_psi[tid * 4 + 2] = make_float2(hv.z * inv, 0.f);
    s_psi[tid * 4 + 3] = make_float2(hv.w * inv, 0.f);
    __syncthreads();

    // --- 4 layers x 10 single-qubit gates + permutation ---
    for (int l = 0; l < NL; ++l) {
        for (int w = 0; w < NQ; ++w) {
            const float* g = s_g[l * NQ + w];
            const float m00r = g[0], m00i = g[1], m01r = g[2], m01i = g[3];
            const float m10r = g[4], m10i = g[5], m11r = g[6], m11i = g[7];
            const int sh = NQ - 1 - w;           // log2(right)
            const int right = 1 << sh;
            #pragma unroll
            for (int pp = 0; pp < 2; ++pp) {
                const int p  = tid + pp * 256;   // pair id, 512 pairs total
                const int i0 = ((p >> sh) << (sh + 1)) | (p & (right - 1));
                const int i1 = i0 + right;
                const float2 a = s_psi[i0];
                const float2 b = s_psi[i1];
                float2 n0, n1;
                n0.x = m00r * a.x - m00i * a.y + m01r * b.x - m01i * b.y;
                n0.y = m00r * a.y + m00i * a.x + m01r * b.y + m01i * b.x;
                n1.x = m10r * a.x - m10i * a.y + m11r * b.x - m11i * b.y;
                n1.y = m10r * a.y + m10i * a.x + m11r * b.y + m11i * b.x;
                s_psi[i0] = n0;
                s_psi[i1] = n1;
            }
            __syncthreads();
        }
        // permutation: new[i] = old[perm(i)]  (gather to regs, then write back)
        const int base = tid * 4;
        const float2 t0 = s_psi[sqe_perm_idx(l, base + 0)];
        const float2 t1 = s_psi[sqe_perm_idx(l, base + 1)];
        const float2 t2 = s_psi[sqe_perm_idx(l, base + 2)];
        const float2 t3 = s_psi[sqe_perm_idx(l, base + 3)];
        __syncthreads();
        s_psi[base + 0] = t0;
        s_psi[base + 1] = t1;
        s_psi[base + 2] = t2;
        s_psi[base + 3] = t3;
        __syncthreads();
    }

    // --- expvals[q] = sum_i |psi_i|^2 * (1 - 2*bit_q(i)) ---
    float eq[NQ];
    #pragma unroll
    for (int q = 0; q < NQ; ++q) eq[q] = 0.f;
    #pragma unroll
    for (int j = 0; j < 4; ++j) {
        const int i = tid * 4 + j;
        const float2 v = s_psi[i];
        const float pr = v.x * v.x + v.y * v.y;
        #pragma unroll
        for (int q = 0; q < NQ; ++q)
            eq[q] += ((i >> (NQ - 1 - q)) & 1) ? -pr : pr;
    }
    #pragma unroll
    for (int q = 0; q < NQ; ++q) {
        #pragma unroll
        for (int off = 16; off; off >>= 1) eq[q] += __shfl_xor(eq[q], off, 32);
    }
    if (lane == 0) {
        #pragma unroll
        for (int q = 0; q < NQ; ++q) s_wq[wv][q] = eq[q];
    }
    __syncthreads();
    if (tid < NQ) {
        float t = 0.f;
        #pragma unroll
        for (int i = 0; i < 8; ++i) t += s_wq[i][tid];
        s_exp[tid] = t;
    }
    __syncthreads();

    // --- out = expvals @ W_out^T + b_out ---
    if (tid < PREDN) {
        float acc = b_out[tid];
        #pragma unroll
        for (int q = 0; q < NQ; ++q) acc += s_exp[q] * W_out[tid * NQ + q];
        out[(size_t)brow * PREDN + tid] = acc;
    }
}

// ---------------------------------------------------------------------------
// Launch: inputs in setup_inputs() order: x, W_in, b_in, q_weights, W_out, b_out
// Workspace holds h (4096*1024 f32 = 16 MB); stays L2-resident (192 MB L2).
// ---------------------------------------------------------------------------
extern "C" void kernel_launch(void* const* d_in, const int* in_sizes, int n_in,
                              void* d_out, int out_size, void* d_ws, size_t ws_size,
                              hipStream_t stream)
{
    const float* x     = (const float*)d_in[0];
    const float* W_in  = (const float*)d_in[1];
    const float* b_in  = (const float*)d_in[2];
    const float* qw    = (const float*)d_in[3];
    const float* W_out = (const float*)d_in[4];
    const float* b_out = (const float*)d_in[5];
    float* out = (float*)d_out;
    float* h   = (float*)d_ws;

    // 128 M-macrotiles * 32 N-macrotiles = 4096 waves, 8 waves/block -> 512 blocks
    sqe_gemm_in<<<(BSZ / 32) * (QDIM / 32) / 8, 256, 0, stream>>>(x, W_in, b_in, h);

    // one block per batch row
    sqe_circuit<<<BSZ, 256, 0, stream>>>(h, qw, W_out, b_out, out);
}